// GCN_58789512348186
// MI455X (gfx1250) — compile-verified
//
#include <hip/hip_runtime.h>
#include <hip/hip_bf16.h>

typedef __attribute__((ext_vector_type(2))) float v2f;
typedef __attribute__((ext_vector_type(8))) float v8f;

// ---------------------------------------------------------------------------
// Build CSR row_ptr from sorted COO rows (int64) via per-node binary search.
// row_ptr[i] = lower_bound(rows, i); row_ptr[n_rows] = n_edges.
// ---------------------------------------------------------------------------
__global__ void gcn_build_row_ptr(const long long* __restrict__ rows,
                                  int* __restrict__ row_ptr,
                                  int n_rows, int n_edges) {
    int i = blockIdx.x * blockDim.x + threadIdx.x;
    if (i > n_rows) return;
    int lo = 0, hi = n_edges;
    long long key = (long long)i;
    while (lo < hi) {
        int mid = (lo + hi) >> 1;
        if (rows[mid] < key) lo = mid + 1; else hi = mid;
    }
    row_ptr[i] = lo;
}

// ---------------------------------------------------------------------------
// Dense GEMM C[M,N] = A[M,K] * B[K,N] (row major) using V_WMMA_F32_16X16X4_F32.
// One wave computes a 16-row strip across all N (N/16 tiles), reusing the A
// fragment for every B tile. fp32 in / fp32 accumulate (exact vs reference).
//
// ISA f32 A-frag layout (16x4): lanes 0-15: M=lane, VGPR0=K0,VGPR1=K1;
//                               lanes 16-31: M=lane-16, VGPR0=K2,VGPR1=K3.
// B-frag (4x16) mirrors with N on lanes. C/D: VGPR r = row r (lanes 0-15) /
// row r+8 (lanes 16-31), col = lane&15.
// ---------------------------------------------------------------------------
template <int K, int N>
__global__ void gcn_gemm_wmma_f32(const float* __restrict__ A,
                                  const float* __restrict__ B,
                                  float* __restrict__ C, int M) {
    constexpr int NT = N / 16;
    const int wave = (int)((blockIdx.x * blockDim.x + threadIdx.x) >> 5);
    const int lane = (int)(threadIdx.x & 31);
    const int m0 = wave * 16;
    if (m0 >= M) return;                    // wave-uniform: EXEC stays all-ones

    const int half = lane >> 4;             // 0 or 1 (selects K pair)
    const int l    = lane & 15;

    v8f acc[NT];
#pragma unroll
    for (int n = 0; n < NT; ++n) acc[n] = (v8f){0.f,0.f,0.f,0.f,0.f,0.f,0.f,0.f};

    int arow_i = m0 + l; if (arow_i > M - 1) arow_i = M - 1;   // clamp (safe load)
    const float* arow = A + (size_t)arow_i * K + half * 2;

    for (int k0 = 0; k0 < K; k0 += 4) {
        const v2f a = *(const v2f*)(arow + k0);                // 8B contiguous
        const float* brow0 = B + (size_t)(k0 + half * 2) * N + l;
        const float* brow1 = brow0 + N;
#pragma unroll
        for (int n = 0; n < NT; ++n) {
            v2f b;
            b.x = brow0[n * 16];
            b.y = brow1[n * 16];
            acc[n] = __builtin_amdgcn_wmma_f32_16x16x4_f32(
                false, a, false, b, (short)0, acc[n], false, false);
        }
    }

#pragma unroll
    for (int n = 0; n < NT; ++n) {
#pragma unroll
        for (int r = 0; r < 8; ++r) {
            const int row = m0 + r + half * 8;
            if (row < M)
                C[(size_t)row * N + n * 16 + l] = acc[n][r];
        }
    }
}

// ---------------------------------------------------------------------------
// SpMM + bias + ReLU, F=128: one wave per output row, float4 per lane.
// out[i,:] = relu( sum_{e in row i} vals[e] * dense[cols[e],:] + bias )
// Gather rows live in L2 (51 MB < 192 MB) -> coalesced 512B per edge.
// ---------------------------------------------------------------------------
__global__ void gcn_spmm_relu_f128(const float* __restrict__ vals,
                                   const long long* __restrict__ cols,
                                   const int* __restrict__ row_ptr,
                                   const float* __restrict__ dense,
                                   const float* __restrict__ bias,
                                   float* __restrict__ out, int n_rows) {
    const int row  = (int)((blockIdx.x * blockDim.x + threadIdx.x) >> 5);
    const int lane = (int)(threadIdx.x & 31);
    if (row >= n_rows) return;
    const int e0 = row_ptr[row], e1 = row_ptr[row + 1];

    float4 acc = make_float4(0.f, 0.f, 0.f, 0.f);
    for (int e = e0; e < e1; ++e) {
        const float     v = vals[e];
        const long long c = cols[e];
        const float4 g = *(const float4*)(dense + (size_t)c * 128 + lane * 4);
        acc.x = fmaf(v, g.x, acc.x);
        acc.y = fmaf(v, g.y, acc.y);
        acc.z = fmaf(v, g.z, acc.z);
        acc.w = fmaf(v, g.w, acc.w);
    }
    const float4 b = *(const float4*)(bias + lane * 4);
    float4 r;
    r.x = fmaxf(acc.x + b.x, 0.f);
    r.y = fmaxf(acc.y + b.y, 0.f);
    r.z = fmaxf(acc.z + b.z, 0.f);
    r.w = fmaxf(acc.w + b.w, 0.f);
    *(float4*)(out + (size_t)row * 128 + lane * 4) = r;
}

// Same, F=64: float2 per lane.
__global__ void gcn_spmm_relu_f64(const float* __restrict__ vals,
                                  const long long* __restrict__ cols,
                                  const int* __restrict__ row_ptr,
                                  const float* __restrict__ dense,
                                  const float* __restrict__ bias,
                                  float* __restrict__ out, int n_rows) {
    const int row  = (int)((blockIdx.x * blockDim.x + threadIdx.x) >> 5);
    const int lane = (int)(threadIdx.x & 31);
    if (row >= n_rows) return;
    const int e0 = row_ptr[row], e1 = row_ptr[row + 1];

    float2 acc = make_float2(0.f, 0.f);
    for (int e = e0; e < e1; ++e) {
        const float     v = vals[e];
        const long long c = cols[e];
        const float2 g = *(const float2*)(dense + (size_t)c * 64 + lane * 2);
        acc.x = fmaf(v, g.x, acc.x);
        acc.y = fmaf(v, g.y, acc.y);
    }
    const float2 b = *(const float2*)(bias + lane * 2);
    float2 r;
    r.x = fmaxf(acc.x + b.x, 0.f);
    r.y = fmaxf(acc.y + b.y, 0.f);
    *(float2*)(out + (size_t)row * 64 + lane * 2) = r;
}

// ---------------------------------------------------------------------------
// Launch: row_ptr -> GEMM1 (WMMA) -> SpMM1+ReLU -> GEMM2 (WMMA) -> SpMM2+ReLU
// ws layout: [row_ptr | H1 (reused as H2) | h]  peak ~103 MB.
// ---------------------------------------------------------------------------
extern "C" void kernel_launch(void* const* d_in, const int* in_sizes, int n_in,
                              void* d_out, int out_size, void* d_ws, size_t ws_size,
                              hipStream_t stream) {
    const float*     x     = (const float*)d_in[0];
    const float*     W1    = (const float*)d_in[1];
    const float*     b1    = (const float*)d_in[2];
    const float*     W2    = (const float*)d_in[3];
    const float*     b2    = (const float*)d_in[4];
    const float*     evals = (const float*)d_in[5];
    const long long* erows = (const long long*)d_in[6];
    const long long* ecols = (const long long*)d_in[7];

    constexpr int NFEAT = 256, NHID = 128, NHID2 = 64;
    const int n_nodes = in_sizes[0] / NFEAT;
    const int n_edges = in_sizes[5];

    char* ws = (char*)d_ws;
    int* row_ptr = (int*)ws;
    size_t off = (((size_t)(n_nodes + 1) * sizeof(int)) + 255u) & ~(size_t)255u;
    float* H1 = (float*)(ws + off);                 // [n_nodes, 128]
    float* h  = H1 + (size_t)n_nodes * NHID;        // [n_nodes, 128]
    float* H2 = H1;                                 // reuse: H1 dead after SpMM1

    // 1) CSR row pointers from sorted COO rows
    {
        int threads = 256;
        int blocks = (n_nodes + 1 + threads - 1) / threads;
        gcn_build_row_ptr<<<blocks, threads, 0, stream>>>(erows, row_ptr,
                                                          n_nodes, n_edges);
    }
    // 2) H1 = x @ W1  (fp32 WMMA)
    {
        int waves = (n_nodes + 15) / 16;
        int threads = 256;                          // 8 waves / block
        int blocks = (waves * 32 + threads - 1) / threads;
        gcn_gemm_wmma_f32<NFEAT, NHID><<<blocks, threads, 0, stream>>>(
            x, W1, H1, n_nodes);
    }
    // 3) h = relu(spmm(H1) + b1)
    {
        int threads = 256;                          // 8 rows / block
        int blocks = (n_nodes * 32 + threads - 1) / threads;
        gcn_spmm_relu_f128<<<blocks, threads, 0, stream>>>(
            evals, ecols, row_ptr, H1, b1, h, n_nodes);
    }
    // 4) H2 = h @ W2  (fp32 WMMA)
    {
        int waves = (n_nodes + 15) / 16;
        int threads = 256;
        int blocks = (waves * 32 + threads - 1) / threads;
        gcn_gemm_wmma_f32<NHID, NHID2><<<blocks, threads, 0, stream>>>(
            h, W2, H2, n_nodes);
    }
    // 5) out = relu(spmm(H2) + b2)
    {
        int threads = 256;
        int blocks = (n_nodes * 32 + threads - 1) / threads;
        gcn_spmm_relu_f64<<<blocks, threads, 0, stream>>>(
            evals, ecols, row_ptr, H2, b2, (float*)d_out, n_nodes);
    }
}